// CNF_2layer_52931176956418
// MI455X (gfx1250) — compile-verified
//
#include <hip/hip_runtime.h>
#include <hip/hip_bf16.h>

// Problem constants (from reference): B=16384, D=512
#define BATCH 16384
#define DD    512          // D
#define HH    1024         // 2*D
#define K1    513          // D+1   (W1 row stride)
#define K2    1025         // 2*D+1 (W2 row stride)
#define BM    32           // batch rows per workgroup
#define HSTR  1032         // LDS h row stride in floats (1024 + 8 pad)

typedef float v2f __attribute__((ext_vector_type(2)));
typedef float v8f __attribute__((ext_vector_type(8)));

static __device__ __forceinline__ v2f load2g(const float* __restrict__ p) {
    // possibly only 4-byte aligned (odd row strides 513/1025)
    v2f r;
    __builtin_memcpy(&r, p, sizeof(r));
    return r;
}

static __device__ __forceinline__ v8f wmma_f32(v2f a, v2f b, v8f c) {
    // D = A(16x4) x B(4x16) + C(16x16), fp32 matrix core path on gfx1250
    return __builtin_amdgcn_wmma_f32_16x16x4_f32(false, a, false, b,
                                                 (short)0, c, false, false);
}

static __device__ __forceinline__ v8f splat8(float v) {
    v8f r = {v, v, v, v, v, v, v, v};
    return r;
}

// ---------------------------------------------------------------------------
// Tiny helper: c[j] = sum_k W2[k,j] * W1[j,k],  j in [0, 2D)
// ---------------------------------------------------------------------------
__global__ void cnf_c_kernel(const float* __restrict__ W1,
                             const float* __restrict__ W2,
                             float* __restrict__ c) {
    int j = blockIdx.x * blockDim.x + threadIdx.x;
    if (j >= HH) return;
    float acc = 0.f;
    const float* w1row = W1 + (size_t)j * K1;
    for (int k = 0; k < DD; ++k)
        acc = fmaf(W2[(size_t)k * K2 + j], w1row[k], acc);
    c[j] = acc;
}

// ---------------------------------------------------------------------------
// Fused CNF kernel: GEMM1 + ReLU + trace + GEMM2, one workgroup per 32 rows.
// 256 threads = 8 wave32s.
//   GEMM1: wave w -> h columns [w*128, w*128+128)
//   GEMM2: wave w -> dz columns [w*64,  w*64+64)
// All fragment loads in the K-loops use ONE running base pointer per matrix
// plus compile-time byte offsets (tile stride is constant), so the loop body
// is loads + WMMAs with a single pointer bump.
// ---------------------------------------------------------------------------
__global__ __launch_bounds__(256, 1)
void cnf_fused_kernel(const float* __restrict__ tptr,
                      const float* __restrict__ z,
                      const float* __restrict__ W1,
                      const float* __restrict__ b1,
                      const float* __restrict__ W2,
                      const float* __restrict__ b2,
                      const float* __restrict__ cws,
                      float* __restrict__ out,   // [B*D] dz_dt, then [B] dlogp
                      int compute_c_local) {
    extern __shared__ float smem[];
    float* hsh  = smem;                  // BM x HSTR  (post-ReLU h tile)
    float* csh  = smem + BM * HSTR;      // HH         (c vector)
    float* trsh = csh + HH;              // BM         (per-row trace)

    const int tid  = threadIdx.x;
    const int wave = tid >> 5;
    const int lane = tid & 31;
    const int l16  = lane & 15;
    const int half = lane >> 4;          // 0: K pair {0,1} / M 0..7, 1: {2,3} / M 8..15
    const int rowbase = blockIdx.x * BM;
    const float tval = tptr[0];

    if (tid < BM) trsh[tid] = 0.f;

    if (!compute_c_local) {
        for (int i = tid; i < HH; i += 256) csh[i] = cws[i];
    } else {
        // fallback: compute c cooperatively (W2 reads coalesce across lanes)
        for (int g = 0; g < 4; ++g) {
            int j = tid + 256 * g;
            float acc = 0.f;
            const float* w1row = W1 + (size_t)j * K1;
            for (int k = 0; k < DD; ++k)
                acc = fmaf(W2[(size_t)k * K2 + j], w1row[k], acc);
            csh[j] = acc;
        }
    }
    __syncthreads();

    // =============================== GEMM1 ================================
    {
        const int jb = wave * 128;

        // Pre-load bias + t-column term into the accumulators: for output
        // column j it is t*W1[j][D] + b1[j], identical for every row (= every
        // accumulator component of a lane).
        v8f acc[2][8];
        #pragma unroll
        for (int n = 0; n < 8; ++n) {
            const int j = jb + n * 16 + l16;
            const float addv = fmaf(tval, W1[(size_t)j * K1 + DD], b1[j]);
            acc[0][n] = splat8(addv);
            acc[1][n] = splat8(addv);
        }

        // Running base pointers; tile strides are compile-time constants so
        // they fold into the 24-bit instruction offsets.
        const float* za = z + (size_t)(rowbase + l16) * DD + 2 * half;
        const float* wb = W1 + (size_t)(jb + l16) * K1 + 2 * half;

        #pragma unroll 2
        for (int k = 0; k < DD; k += 4) {
            v2f a0 = *(const v2f*)(za);              // rows 0..15
            v2f a1 = *(const v2f*)(za + 16 * DD);    // rows 16..31 (+32768 B)
            #pragma unroll
            for (int n = 0; n < 8; ++n) {
                v2f b = load2g(wb + (size_t)n * 16 * K1);  // +n*32832 B
                acc[0][n] = wmma_f32(a0, b, acc[0][n]);
                acc[1][n] = wmma_f32(a1, b, acc[1][n]);
            }
            za += 4;
            wb += 4;
        }

        // epilogue: ReLU -> LDS, masked-c trace accumulation.
        // FULLY unrolled: n indexes the register array acc[][].
        float tr[2][8];
        #pragma unroll
        for (int mb = 0; mb < 2; ++mb)
            #pragma unroll
            for (int r = 0; r < 8; ++r) tr[mb][r] = 0.f;

        #pragma unroll
        for (int n = 0; n < 8; ++n) {
            const int j = jb + n * 16 + l16;
            const float cj = csh[j];
            #pragma unroll
            for (int mb = 0; mb < 2; ++mb) {
                #pragma unroll
                for (int r = 0; r < 8; ++r) {
                    const float hp = acc[mb][n][r];   // already includes bias+t
                    hsh[(size_t)(mb * 16 + r + 8 * half) * HSTR + j] =
                        fmaxf(hp, 0.f);
                    tr[mb][r] += (hp > 0.f) ? cj : 0.f;
                }
            }
        }

        // reduce trace over the 16 lanes of each half, then LDS float atomics
        #pragma unroll
        for (int mb = 0; mb < 2; ++mb) {
            #pragma unroll
            for (int r = 0; r < 8; ++r) {
                float s = tr[mb][r];
                s += __shfl_xor(s, 1, 32);
                s += __shfl_xor(s, 2, 32);
                s += __shfl_xor(s, 4, 32);
                s += __shfl_xor(s, 8, 32);
                if (l16 == 0)
                    atomicAdd(&trsh[mb * 16 + r + 8 * half], s);
            }
        }
    }
    __syncthreads();

    // dlogp_z/dt = -trace
    if (tid < BM)
        out[(size_t)BATCH * DD + rowbase + tid] = -trsh[tid];

    // =============================== GEMM2 ================================
    {
        const int jb2 = wave * 64;

        v8f acc2[2][4];
        #pragma unroll
        for (int n = 0; n < 4; ++n) {
            const int j = jb2 + n * 16 + l16;
            const float addv = fmaf(tval, W2[(size_t)j * K2 + HH], b2[j]);
            acc2[0][n] = splat8(addv);
            acc2[1][n] = splat8(addv);
        }

        const float* ha0 = hsh + (size_t)l16 * HSTR + 2 * half;        // rows 0..15
        const float* ha1 = hsh + (size_t)(16 + l16) * HSTR + 2 * half; // rows 16..31
        const float* wb2 = W2 + (size_t)(jb2 + l16) * K2 + 2 * half;

        #pragma unroll 2
        for (int k = 0; k < HH; k += 4) {
            v2f a0 = *(const v2f*)(ha0);   // ds_load_b64 (8B aligned)
            v2f a1 = *(const v2f*)(ha1);
            #pragma unroll
            for (int n = 0; n < 4; ++n) {
                v2f b = load2g(wb2 + (size_t)n * 16 * K2);  // +n*65600 B
                acc2[0][n] = wmma_f32(a0, b, acc2[0][n]);
                acc2[1][n] = wmma_f32(a1, b, acc2[1][n]);
            }
            ha0 += 4;
            ha1 += 4;
            wb2 += 4;
        }

        // FULLY unrolled store epilogue (n indexes acc2[][])
        #pragma unroll
        for (int n = 0; n < 4; ++n) {
            const int j = jb2 + n * 16 + l16;
            #pragma unroll
            for (int mb = 0; mb < 2; ++mb) {
                #pragma unroll
                for (int r = 0; r < 8; ++r) {
                    const int row = rowbase + mb * 16 + r + 8 * half;
                    out[(size_t)row * DD + j] = acc2[mb][n][r];
                }
            }
        }
    }
}

// ---------------------------------------------------------------------------
extern "C" void kernel_launch(void* const* d_in, const int* in_sizes, int n_in,
                              void* d_out, int out_size, void* d_ws, size_t ws_size,
                              hipStream_t stream) {
    (void)in_sizes; (void)n_in; (void)out_size;
    const float* t  = (const float*)d_in[0];
    const float* z  = (const float*)d_in[1];
    // d_in[2] = logp_z (zeros, unused by the math)
    const float* W1 = (const float*)d_in[3];
    const float* b1 = (const float*)d_in[4];
    const float* W2 = (const float*)d_in[5];
    const float* b2 = (const float*)d_in[6];
    float* out = (float*)d_out;

    const int use_ws = (ws_size >= (size_t)HH * sizeof(float)) ? 1 : 0;
    if (use_ws) {
        cnf_c_kernel<<<dim3(HH / 256), dim3(256), 0, stream>>>(W1, W2, (float*)d_ws);
    }

    const size_t shmem = ((size_t)BM * HSTR + HH + BM) * sizeof(float); // ~136 KB
    (void)hipFuncSetAttribute(reinterpret_cast<const void*>(cnf_fused_kernel),
                              hipFuncAttributeMaxDynamicSharedMemorySize,
                              (int)shmem);
    cnf_fused_kernel<<<dim3(BATCH / BM), dim3(256), shmem, stream>>>(
        t, z, W1, b1, W2, b2, (const float*)d_ws, out, use_ws ? 0 : 1);
}